// LSTMfeature_74474732913017
// MI455X (gfx1250) — compile-verified
//
#include <hip/hip_runtime.h>

#define B_ 256
#define T_ 2048
#define F_ 64
#define H_ 8
#define O_ 64
#define G_ 32   // 4*H

typedef float v2f __attribute__((ext_vector_type(2)));
typedef float v8f __attribute__((ext_vector_type(8)));

// ---------------------------------------------------------------------------
// Kernel A: xg[branch][r][g] = sum_k x[r][k]*W_ih[g][k] + b_ih[g] + b_hh[g]
// r = b*T + t (row-major [B,T] flatten). One wave handles a 16-row x 32-gate
// tile via V_WMMA_F32_16X16X4_F32 (fp32 in/out, bit-compatible with ref).
// ---------------------------------------------------------------------------
__global__ void xg_gemm_kernel(const float* __restrict__ seqX,
                               const float* __restrict__ seqY,
                               const float* __restrict__ W_ih,
                               const float* __restrict__ b_ih,
                               const float* __restrict__ b_hh,
                               float* __restrict__ xg) {
    const int wave = (blockIdx.x * blockDim.x + threadIdx.x) >> 5;
    const int lane = threadIdx.x & 31;
    const int tilesPerBranch = (B_ * T_) / 16;          // 32768
    const int branch = wave / tilesPerBranch;
    const int tile   = wave - branch * tilesPerBranch;
    const int r0     = tile * 16;

    const int m    = lane & 15;          // A row / B,D column within tile
    const int koff = (lane >> 4) << 1;   // lanes 0-15: K=0,1 ; lanes 16-31: K=2,3

    const float* x  = (branch ? seqY : seqX) + (size_t)(r0 + m) * F_;
    const float* w0 = W_ih + (size_t)m        * F_;     // gate n = m
    const float* w1 = W_ih + (size_t)(16 + m) * F_;     // gate n = 16+m

    v8f acc0 = {};
    v8f acc1 = {};
    #pragma unroll
    for (int k = 0; k < F_; k += 4) {
        v2f a, b0, b1;
        a.x  = x[k + koff];       a.y  = x[k + koff + 1];
        b0.x = w0[k + koff];      b0.y = w0[k + koff + 1];
        b1.x = w1[k + koff];      b1.y = w1[k + koff + 1];
        // (neg_a, A, neg_b, B, c_mod, C, reuse_a, reuse_b)
        acc0 = __builtin_amdgcn_wmma_f32_16x16x4_f32(false, a, false, b0,
                                                     (short)0, acc0, false, false);
        acc1 = __builtin_amdgcn_wmma_f32_16x16x4_f32(false, a, false, b1,
                                                     (short)0, acc1, false, false);
    }

    const float bias0 = b_ih[m]      + b_hh[m];
    const float bias1 = b_ih[16 + m] + b_hh[16 + m];
    const int   mrow  = (lane >> 4) << 3;               // D rows: i + (lane>=16 ? 8 : 0)
    float* dst = xg + (size_t)branch * ((size_t)B_ * T_ * G_);
    #pragma unroll
    for (int i = 0; i < 8; ++i) {
        const size_t r = (size_t)(r0 + mrow + i);
        dst[r * G_ + m]      = acc0[i] + bias0;
        dst[r * G_ + 16 + m] = acc1[i] + bias1;
    }
}

// ---------------------------------------------------------------------------
// Kernel B: sequential LSTM scan + fused masked hidden2obs projection.
// One wave per (branch, sequence); lane g in [0,32) owns gate g (i/f/g/o
// order), lanes 0..7 own the cell state for hidden unit j = lane.
// ---------------------------------------------------------------------------
__device__ __forceinline__ float fast_sigmoid(float x) {
    return __builtin_amdgcn_rcpf(1.0f + __expf(-x));
}
__device__ __forceinline__ float fast_tanh(float x) {
    const float e = __expf(-2.0f * x);
    return (1.0f - e) * __builtin_amdgcn_rcpf(1.0f + e);
}

__global__ void lstm_scan_kernel(const float* __restrict__ xg,
                                 const int* __restrict__ lenX,
                                 const int* __restrict__ lenY,
                                 const float* __restrict__ h0,
                                 const float* __restrict__ c0,
                                 const float* __restrict__ W_hh,
                                 const float* __restrict__ W_out,
                                 const float* __restrict__ b_out,
                                 float* __restrict__ out) {
    const int wave   = (blockIdx.x * blockDim.x + threadIdx.x) >> 5;
    const int lane   = threadIdx.x & 31;
    const int branch = wave / B_;
    const int b      = wave - branch * B_;
    const int len    = branch ? lenY[b] : lenX[b];

    // Recurrent weights for this lane's gate: W_hh[lane][0..7]
    float whh[H_];
    #pragma unroll
    for (int j = 0; j < H_; ++j) whh[j] = W_hh[lane * H_ + j];

    // Output projection weights: lane owns outputs o = lane and o = lane+32
    float wo0[H_], wo1[H_];
    #pragma unroll
    for (int j = 0; j < H_; ++j) {
        wo0[j] = W_out[lane * H_ + j];
        wo1[j] = W_out[(lane + 32) * H_ + j];
    }
    const float bo0 = b_out[lane];
    const float bo1 = b_out[lane + 32];

    // State: h broadcast to all lanes; c held by lanes 0..7 (unit j = lane)
    float h[H_];
    #pragma unroll
    for (int j = 0; j < H_; ++j) h[j] = h0[b * H_ + j];
    float c = (lane < H_) ? c0[b * H_ + lane] : 0.0f;

    const float* xgp  = xg  + (size_t)branch * ((size_t)B_ * T_ * G_)
                            + (size_t)b * T_ * G_;
    float*       outp = out + (size_t)branch * ((size_t)B_ * T_ * O_)
                            + (size_t)b * T_ * O_;

    const bool isTanhLane = (lane >= 16) && (lane < 24);   // 'g' gates

    for (int t = 0; t < T_; ++t) {
        __builtin_prefetch(xgp + (t + 8) * G_ + lane, 0, 0);  // global_prefetch_b8

        // gate pre-activation: xg + h @ W_hh^T  (8 FMAs on the critical path)
        float acc = xgp[t * G_ + lane];
        #pragma unroll
        for (int j = 0; j < H_; ++j) acc += h[j] * whh[j];

        const float act = isTanhLane ? fast_tanh(acc) : fast_sigmoid(acc);

        // Gather i/f/g/o per hidden unit (valid in lanes 0..7; rest harmless)
        const float a_f = __shfl(act, (lane + 8)  & 31, 32);
        const float a_g = __shfl(act, (lane + 16) & 31, 32);
        const float a_o = __shfl(act, (lane + 24) & 31, 32);
        c = a_f * c + act * a_g;                 // act == sigmoid(i) in lanes 0..7
        const float hv = a_o * fast_tanh(c);

        // Re-broadcast h[0..7] to all lanes
        #pragma unroll
        for (int j = 0; j < H_; ++j) h[j] = __shfl(hv, j, 32);

        // Fused hidden2obs with pad mask (masked rows -> b_out, as in ref)
        const float msk = (t < len) ? 1.0f : 0.0f;
        float o0 = 0.0f, o1 = 0.0f;
        #pragma unroll
        for (int j = 0; j < H_; ++j) {
            o0 += h[j] * wo0[j];
            o1 += h[j] * wo1[j];
        }
        outp[t * O_ + lane]      = msk * o0 + bo0;
        outp[t * O_ + 32 + lane] = msk * o1 + bo1;
    }
}

// ---------------------------------------------------------------------------
extern "C" void kernel_launch(void* const* d_in, const int* in_sizes, int n_in,
                              void* d_out, int out_size, void* d_ws, size_t ws_size,
                              hipStream_t stream) {
    (void)in_sizes; (void)n_in; (void)out_size; (void)ws_size;
    const float* seqX  = (const float*)d_in[0];
    const float* seqY  = (const float*)d_in[1];
    const int*   lenX  = (const int*)  d_in[2];
    const int*   lenY  = (const int*)  d_in[3];
    const float* h0    = (const float*)d_in[4];
    const float* c0    = (const float*)d_in[5];
    const float* W_ih  = (const float*)d_in[6];
    const float* W_hh  = (const float*)d_in[7];
    const float* b_ih  = (const float*)d_in[8];
    const float* b_hh  = (const float*)d_in[9];
    const float* W_out = (const float*)d_in[10];
    const float* b_out = (const float*)d_in[11];
    float* out = (float*)d_out;
    float* xg  = (float*)d_ws;   // 2 * B * T * 32 floats = 128 MB scratch

    // Kernel A: 2 branches * 32768 tiles = 65536 waves, 8 waves (256 thr)/block
    const int wavesA = 2 * ((B_ * T_) / 16);
    xg_gemm_kernel<<<wavesA / 8, 256, 0, stream>>>(seqX, seqY, W_ih, b_ih, b_hh, xg);

    // Kernel B: 2 * 256 = 512 waves, 8 waves/block (same-stream ordering
    // guarantees xg is complete before the scan consumes it)
    lstm_scan_kernel<<<(2 * B_) / 8, 256, 0, stream>>>(xg, lenX, lenY, h0, c0,
                                                       W_hh, W_out, b_out, out);
}